// DistanceNorm_4801773437268
// MI455X (gfx1250) — compile-verified
//
#include <hip/hip_runtime.h>
#include <math.h>

typedef __attribute__((ext_vector_type(2))) float v2f;
typedef __attribute__((ext_vector_type(8))) float v8f;

#define Bb 64
#define Ll 2048
#define Mm 256
#define SPLIT 16
#define ROWS_PER_SPLIT (Ll / SPLIT) // 128
#define GROWS 32                    // rows staged per gather block

// ---------------------------------------------------------------------------
// Kernel 1: px partial column sums via WMMA f32 16x16x4 with A = ones.
// D[i][j] = sum_k B[k][j] + C[i][j]; row 0 of D (lanes 0..15, VGPR0 of C/D)
// holds exact f32 column sums. Each wave owns a 16-column strip; grid splits
// rows 16 ways for parallelism; partials are reduced deterministically later.
// ---------------------------------------------------------------------------
__global__ __launch_bounds__(256) void kSum(const float* __restrict__ x,
                                            float* __restrict__ part) {
    const int split = blockIdx.x;
    const int b     = blockIdx.y;
    const int tid   = threadIdx.x;
    const int wave  = tid >> 5;
    const int lane  = tid & 31;
    const int half  = lane >> 4;   // which pair of K-rows this lane feeds
    const int colin = lane & 15;   // N within strip
    const float* xb = x + (size_t)b * Ll * Mm;
    const int row0  = split * ROWS_PER_SPLIT;

    v2f a; a.x = 1.0f; a.y = 1.0f; // A matrix: all ones

    for (int si = 0; si < 2; ++si) {
        const int strip = wave + si * 8;       // 8 waves x 2 = 16 strips
        const int col   = strip * 16 + colin;
        v8f acc = {};
        for (int it = 0; it < ROWS_PER_SPLIT; it += 4) {
            const int rr = row0 + it + half * 2;
            v2f bm;
            bm.x = xb[(size_t)rr * Mm + col];
            bm.y = xb[(size_t)(rr + 1) * Mm + col];
            acc = __builtin_amdgcn_wmma_f32_16x16x4_f32(
                false, a, false, bm, (short)0, acc, false, false);
        }
        if (lane < 16)
            part[((size_t)split * Bb + b) * Mm + col] = acc[0];
    }
}

// ---------------------------------------------------------------------------
// Kernel 2: per-batch stats -> fl (floor index into padded axis) and w (frac).
// One block per batch, 256 threads (= M bins), LDS tree reductions.
// ---------------------------------------------------------------------------
__global__ __launch_bounds__(256) void kStats(const float* __restrict__ part,
                                              int* __restrict__ flb,
                                              float* __restrict__ wb) {
    const int b = blockIdx.x;
    const int m = threadIdx.x;
    __shared__ float red[256];

    float p = 0.0f;
    for (int s = 0; s < SPLIT; ++s)
        p += part[((size_t)s * Bb + b) * Mm + m];

    // S = sum(px)
    red[m] = p; __syncthreads();
    for (int o = 128; o > 0; o >>= 1) { if (m < o) red[m] += red[m + o]; __syncthreads(); }
    const float S = red[0]; __syncthreads();

    const float rng = (float)m - 127.0f;  // arange(M) - M/2 + 1

    // mean = sum(pxn * rng)
    red[m] = p * rng; __syncthreads();
    for (int o = 128; o > 0; o >>= 1) { if (m < o) red[m] += red[m + o]; __syncthreads(); }
    const float mean = red[0] / S; __syncthreads();

    // std = sqrt(sum(pxn * (rng - mean)^2))
    const float d = rng - mean;
    red[m] = p * d * d; __syncthreads();
    for (int o = 128; o > 0; o >>= 1) { if (m < o) red[m] += red[m + o]; __syncthreads(); }
    const float stdv = sqrtf(red[0] / S);

    const float ni  = rng * stdv * (1.0f / 25.6f) + mean + 127.0f; // + M/2 - 1
    float idx = ni + 1.0f;                                         // pad shift
    idx = fminf(fmaxf(idx, 0.0f), 257.0f);                         // clip [0, M+1]
    const int f = (int)idx;                                        // truncation
    flb[b * Mm + m] = f;
    wb[b * Mm + m]  = idx - (float)f;
}

// ---------------------------------------------------------------------------
// Kernel 3: gather + lerp. Stage GROWS rows (32 KB) into LDS with async
// global->LDS B128 copies (ASYNCcnt), then each thread produces column tid
// for all staged rows. Output stores fully coalesced. x re-read hits L2.
// ---------------------------------------------------------------------------
__global__ __launch_bounds__(256) void kGather(const float* __restrict__ x,
                                               const int* __restrict__ flb,
                                               const float* __restrict__ wb,
                                               float* __restrict__ out) {
    const int rb  = blockIdx.x;
    const int b   = blockIdx.y;
    const int tid = threadIdx.x;

    __shared__ float tile[GROWS * Mm];
    __shared__ int   sfl[Mm];
    __shared__ float sw[Mm];

    sfl[tid] = flb[b * Mm + tid];
    sw[tid]  = wb[b * Mm + tid];

    const int row0  = rb * GROWS;
    const float* xb = x + ((size_t)b * Ll + row0) * Mm;

    // 256 threads x 8 iters x 16B = 32 KB; 4 rows per iteration, 64 lanes/row
    const int rsub = tid >> 6;
    const int c4   = (tid & 63) * 4;
    for (int it = 0; it < 8; ++it) {
        const int r = it * 4 + rsub;
        unsigned long long gaddr =
            (unsigned long long)(xb + (size_t)r * Mm + c4);
        unsigned int laddr =
            (unsigned int)(unsigned long long)(&tile[r * Mm + c4]);
        asm volatile("global_load_async_to_lds_b128 %0, %1, off"
                     :: "v"(laddr), "v"(gaddr) : "memory");
    }
    asm volatile("s_wait_asynccnt 0x0" ::: "memory");
    __syncthreads();

    const int   f  = sfl[tid];                      // 0..257 (padded axis)
    const float wt = sw[tid];
    const int   ci = (f + 1 > 257) ? 257 : (f + 1); // OOB ceil clamps (w==0)

    float* ob = out + ((size_t)b * Ll + row0) * Mm + tid;
    for (int r = 0; r < GROWS; ++r) {
        const float* rowp = &tile[r * Mm];
        const float v0 = (f >= 1 && f <= Mm) ? rowp[f - 1] : 0.0f;  // pad = 0
        const float v1 = (ci <= Mm) ? rowp[ci - 1] : 0.0f;          // ci >= 1
        ob[(size_t)r * Mm] = v1 * wt + v0 * (1.0f - wt);
    }
}

// ---------------------------------------------------------------------------
extern "C" void kernel_launch(void* const* d_in, const int* in_sizes, int n_in,
                              void* d_out, int out_size, void* d_ws, size_t ws_size,
                              hipStream_t stream) {
    const float* x = (const float*)d_in[0];
    float* out = (float*)d_out;

    char* ws = (char*)d_ws;
    float* part = (float*)ws;                                   // 1 MiB
    int*   flb  = (int*)(ws + (size_t)SPLIT * Bb * Mm * 4);     // 64 KiB
    float* wb   = (float*)(ws + (size_t)SPLIT * Bb * Mm * 4
                              + (size_t)Bb * Mm * 4);           // 64 KiB

    kSum<<<dim3(SPLIT, Bb), 256, 0, stream>>>(x, part);
    kStats<<<Bb, 256, 0, stream>>>(part, flb, wb);
    kGather<<<dim3(Ll / GROWS, Bb), 256, 0, stream>>>(x, flb, wb, out);
}